// IntraAgg_62423054680429
// MI455X (gfx1250) — compile-verified
//
#include <hip/hip_runtime.h>

typedef __attribute__((ext_vector_type(2))) float v2f;
typedef __attribute__((ext_vector_type(4))) float v4f;
typedef __attribute__((ext_vector_type(8))) float v8f;

#define D_FEAT 256
#define SEGS_PER_BLOCK 16
#define CHUNK 256           // edges staged in LDS per iteration (multiple of 4)

// One block = 16 consecutive batch rows (segments). 4 waves x 32 lanes; each
// wave owns a 64-column slice. Segment sums are computed with
// V_WMMA_F32_16X16X4_F32: A = 16x4 {0,1} selection matrix (which segment each
// of 4 edges belongs to), B = 4x16 gathered feature values. Exact f32 math.
//
// Data-movement policy for MI455X: the 102.4 MB feature table is L2-resident
// (192 MB L2), gathers use default (RT) temporal hints; self_feats/output are
// single-touch streams and use non-temporal hints so they don't evict it.
__global__ __launch_bounds__(128) void intra_agg_wmma_kernel(
    const float* __restrict__ features,
    const long long* __restrict__ neigh_idx,
    const long long* __restrict__ segment_ids,
    const float* __restrict__ self_feats,
    float* __restrict__ out,
    int n_edges)
{
    __shared__ int s_start[SEGS_PER_BLOCK + 1];
    __shared__ int s_off[CHUNK];   // byte offset of gathered row (idx * 1024)
    __shared__ int s_seg[CHUNK];   // local segment (0..15) per staged edge, -1 = pad

    const int tid   = threadIdx.x;
    const int lane  = tid & 31;
    const int wave  = tid >> 5;
    const int b0    = blockIdx.x * SEGS_PER_BLOCK;
    const int c     = lane & 15;   // A: row m ; B/C/D: column-within-tile n
    const int kh    = lane >> 4;   // lane-half -> K half for A/B, M offset 8 for C/D
    const int wbase = wave * 64;   // this wave's 64-column slice of D_FEAT

    // Per-lane byte offset of this lane's float4 within a feature row.
    const unsigned colbyte = (unsigned)(wbase + 4 * c) * 4u;
    const char* const fbase = (const char*)features;

    // lower_bound(segment_ids, b0 + t) for t = 0..16 -> segment starts + block range
    if (tid <= SEGS_PER_BLOCK) {
        const long long target = (long long)(b0 + tid);
        int lo = 0, hi = n_edges;
        while (lo < hi) {
            const int mid = (lo + hi) >> 1;
            if (segment_ids[mid] < target) lo = mid + 1; else hi = mid;
        }
        s_start[tid] = lo;
    }
    __syncthreads();

    const int e_begin = s_start[0];
    const int e_end   = s_start[SEGS_PER_BLOCK];

    v8f acc0 = {}, acc1 = {}, acc2 = {}, acc3 = {};   // 4 N-tiles (column-permuted)

    for (int cbase = e_begin; cbase < e_end; cbase += CHUNK) {
        const int cnum = min(CHUNK, e_end - cbase);
        for (int i = tid; i < CHUNK; i += 128) {
            if (i < cnum) {
                s_off[i] = ((int)neigh_idx[cbase + i]) << 10;   // row byte offset
                s_seg[i] = (int)(segment_ids[cbase + i] - (long long)b0);
            } else {
                s_off[i] = 0;      // safe dummy row; masked out by A = 0
                s_seg[i] = -1;
            }
        }
        if (cbase + CHUNK < e_end) {
            __builtin_prefetch(neigh_idx + cbase + CHUNK, 0, 1);   // global_prefetch_b8
        }
        __syncthreads();

        for (int k0 = 0; k0 < cnum; k0 += 4) {
            const int ea = k0 + 2 * kh;       // K = 2*kh   (this lane-half's K pair)
            const int eb = ea + 1;            // K = 2*kh+1
            const int sa = s_seg[ea];
            const int sb = s_seg[eb];
            const unsigned oa = (unsigned)s_off[ea] + colbyte;
            const unsigned ob = (unsigned)s_off[eb] + colbyte;

            // A 16x4: VGPR0 = K=2*kh, VGPR1 = K=2*kh+1, lanes 0..15 hold M=c
            v2f amat;
            amat.x = (sa == c) ? 1.0f : 0.0f;
            amat.y = (sb == c) ? 1.0f : 0.0f;

            // One float4 per edge row covers all 4 N-tiles (component j of the
            // float4 belongs to tile j at N-index c => memory col wbase+4c+j).
            // 32-bit offsets from the uniform base -> SADDR-form global loads.
            const v4f fa = *(const v4f*)(fbase + oa);
            const v4f fb = *(const v4f*)(fbase + ob);

            v2f b;
            b.x = fa[0]; b.y = fb[0];
            acc0 = __builtin_amdgcn_wmma_f32_16x16x4_f32(false, amat, false, b, (short)0, acc0, false, false);
            b.x = fa[1]; b.y = fb[1];
            acc1 = __builtin_amdgcn_wmma_f32_16x16x4_f32(false, amat, false, b, (short)0, acc1, false, false);
            b.x = fa[2]; b.y = fb[2];
            acc2 = __builtin_amdgcn_wmma_f32_16x16x4_f32(false, amat, false, b, (short)0, acc2, false, false);
            b.x = fa[3]; b.y = fb[3];
            acc3 = __builtin_amdgcn_wmma_f32_16x16x4_f32(false, amat, false, b, (short)0, acc3, false, false);
        }
        __syncthreads();
    }

    // Epilogue: C/D layout => VGPR v holds row M = v + 8*kh, N = c.
    // Invert the column permutation: tile j, N=c -> memory column wbase + 4c + j,
    // so packing (acc0..acc3)[v] is exactly the contiguous float4 at wbase+4c.
#pragma unroll
    for (int v = 0; v < 8; ++v) {
        const int mloc  = v + 8 * kh;
        const int row   = b0 + mloc;
        const int count = s_start[mloc + 1] - s_start[mloc];
        const float inv = 1.0f / (float)(count > 0 ? count : 1);

        v4f mean;
        mean[0] = acc0[v] * inv;
        mean[1] = acc1[v] * inv;
        mean[2] = acc2[v] * inv;
        mean[3] = acc3[v] * inv;

        // Single-touch stream: non-temporal load, keep L2 for the feature table.
        const v4f self4 = __builtin_nontemporal_load(
            (const v4f*)(self_feats + (size_t)row * D_FEAT + wbase + 4 * c));
        const v4f diff = self4 - mean;

        float* orow = out + (size_t)row * (2 * D_FEAT);
        __builtin_nontemporal_store(diff, (v4f*)(orow + wbase + 4 * c));          // self - agg
        __builtin_nontemporal_store(mean, (v4f*)(orow + D_FEAT + wbase + 4 * c)); // agg
    }
}

extern "C" void kernel_launch(void* const* d_in, const int* in_sizes, int n_in,
                              void* d_out, int out_size, void* d_ws, size_t ws_size,
                              hipStream_t stream) {
    (void)n_in; (void)out_size; (void)d_ws; (void)ws_size;
    const float*     features    = (const float*)d_in[0];
    const long long* neigh_idx   = (const long long*)d_in[1];   // int64 per reference
    const long long* segment_ids = (const long long*)d_in[2];   // int64, sorted
    const float*     self_feats  = (const float*)d_in[3];
    float*           out         = (float*)d_out;

    const int n_edges = in_sizes[1];
    const int n_batch = in_sizes[3] / D_FEAT;
    const int n_blocks = n_batch / SEGS_PER_BLOCK;   // 16384 / 16 = 1024

    intra_agg_wmma_kernel<<<n_blocks, 128, 0, stream>>>(
        features, neigh_idx, segment_ids, self_feats, out, n_edges);
}